// RecurrentHedgeModel_28552942583919
// MI455X (gfx1250) — compile-verified
//
#include <hip/hip_runtime.h>

typedef __attribute__((ext_vector_type(2))) float v2f;
typedef __attribute__((ext_vector_type(8))) float v8f;

#define STEPS 30
#define SDIM  31   // STEPS + 1
#define ROWS_PER_BLOCK 256  // 8 waves * 32 rows

__global__ __launch_bounds__(256) void hedge_wmma_kernel(
    const float* __restrict__ S,   // (batch, 31)
    const float* __restrict__ W1,  // (30, 2, 16)
    const float* __restrict__ b1,  // (30, 16)
    const float* __restrict__ W2,  // (30, 16, 1)
    const float* __restrict__ b2,  // (30, 1)
    float* __restrict__ out,       // (batch, 30)
    int batch)
{
    const int lane = threadIdx.x & 31;
    const int wave = threadIdx.x >> 5;
    const int l    = lane & 15;
    const bool hi  = lane >= 16;

    const int base = blockIdx.x * ROWS_PER_BLOCK + wave * 32;
    if (base >= batch) return;   // wave-uniform: EXEC stays all-ones for WMMA

    // Each lane owns one batch row for load/store duty:
    // lanes 0..15 -> tile0 rows base+l, lanes 16..31 -> tile1 rows base+16+l
    const int row = base + (hi ? 16 : 0) + l;
    const float* __restrict__ Srow = S   + (long)row * SDIM;
    float*       __restrict__ Orow = out + (long)row * STEPS;

    // deltas: after each step, ALL lanes hold d0 = delta_tile0[l], d1 = delta_tile1[l]
    float d0 = 0.0f, d1 = 0.0f;

#pragma unroll 1
    for (int k = 0; k < STEPS; ++k) {
        // ---- A = W1^T (16x4), M=hidden, K=input-feature; K=2,3 are zero pad ----
        const float* W1k = W1 + k * 32;           // (2,16) block for this step
        float w1r0 = W1k[l];                      // W1[k][0][l]
        float w1r1 = W1k[16 + l];                 // W1[k][1][l]
        v2f A;
        A.x = hi ? 0.0f : w1r0;                   // lanes 0..15: A[M=l][K=0]
        A.y = hi ? 0.0f : w1r1;                   // lanes 0..15: A[M=l][K=1]

        // ---- C = b1 broadcast over batch columns (per-half hidden rows) ----
        v8f C = *(const v8f*)(b1 + k * 16 + (hi ? 8 : 0));   // 32B-aligned

        // ---- layer-2 weights for this half's hidden rows ----
        v8f w2v = *(const v8f*)(W2 + k * 16 + (hi ? 8 : 0)); // 32B-aligned
        float b2v = b2[k];

        // ---- build B matrices (4 x 16batch): row0 = s_k, row1 = delta_{k-1} ----
        float sv = Srow[k];                       // lane<16: tile0 s; lane>=16: tile1 s
        float s1 = __shfl_xor(sv, 16, 32);        // lanes 0..15 now see tile1 s

        v2f B0, B1;
        B0.x = hi ? 0.0f : sv;   B0.y = hi ? 0.0f : d0;
        B1.x = hi ? 0.0f : s1;   B1.y = hi ? 0.0f : d1;

        // ---- layer 1: h^T = relu(W1^T @ inp^T + b1) via f32 WMMA ----
        v8f H0 = __builtin_amdgcn_wmma_f32_16x16x4_f32(
            false, A, false, B0, (short)0, C, false, false);
        v8f H1 = __builtin_amdgcn_wmma_f32_16x16x4_f32(
            false, A, false, B1, (short)0, C, false, false);

        // ---- layer 2: delta = h . w2 + b2 ----
        // lane n holds h[0..7][n], lane n+16 holds h[8..15][n] -> partial dot per half
        float p0 = 0.0f, p1 = 0.0f;
#pragma unroll
        for (int r = 0; r < 8; ++r) {
            p0 = fmaf(fmaxf(H0[r], 0.0f), w2v[r], p0);
            p1 = fmaf(fmaxf(H1[r], 0.0f), w2v[r], p1);
        }
        d0 = p0 + __shfl_xor(p0, 16, 32) + b2v;   // all lanes: delta_tile0[l]
        d1 = p1 + __shfl_xor(p1, 16, 32) + b2v;   // all lanes: delta_tile1[l]

        // ---- store: each lane writes its owned row ----
        Orow[k] = hi ? d1 : d0;
    }
}

extern "C" void kernel_launch(void* const* d_in, const int* in_sizes, int n_in,
                              void* d_out, int out_size, void* d_ws, size_t ws_size,
                              hipStream_t stream) {
    const float* S  = (const float*)d_in[0];
    const float* W1 = (const float*)d_in[1];
    const float* b1 = (const float*)d_in[2];
    const float* W2 = (const float*)d_in[3];
    const float* b2 = (const float*)d_in[4];
    float* out = (float*)d_out;

    const int batch = in_sizes[0] / SDIM;
    const int grid  = (batch + ROWS_PER_BLOCK - 1) / ROWS_PER_BLOCK;

    hedge_wmma_kernel<<<grid, 256, 0, stream>>>(S, W1, b1, W2, b2, out, batch);
}